// RGCN2_combine_losses_4037269258411
// MI455X (gfx1250) — compile-verified
//
#include <hip/hip_runtime.h>

typedef __attribute__((ext_vector_type(2))) float v2f;
typedef __attribute__((ext_vector_type(8))) float v8f;

__device__ __forceinline__ float lrelu(float v) { return v > 0.0f ? v : 0.01f * v; }

// ---------------- utility kernels ----------------

__global__ __launch_bounds__(256) void zero_f32(float* __restrict__ p, long n) {
  long g = (long)blockIdx.x * 256 + threadIdx.x;
  if (g < n) p[g] = 0.0f;
}

__global__ __launch_bounds__(256) void count_edges(const int* __restrict__ idx,
                                                   int* __restrict__ cnt, int E) {
  int g = blockIdx.x * 256 + threadIdx.x;
  if (g < E) atomicAdd(cnt + idx[g], 1);
}

// in-place: buf currently holds int32 counts; write rsqrt(max(cnt,1)) as f32
__global__ __launch_bounds__(256) void deg_to_rinv(float* __restrict__ buf, int n) {
  int g = blockIdx.x * 256 + threadIdx.x;
  if (g < n) {
    int c = ((const int*)buf)[g];
    float d = (c < 1) ? 1.0f : (float)c;
    buf[g] = rsqrtf(d);
  }
}

// tmp[dst[e]][:] += h[src[e]][:] * rinv_src[src[e]]   (32 lanes/edge, float4 each)
__global__ __launch_bounds__(256) void scatter_msgs(
    const int* __restrict__ src, const int* __restrict__ dst,
    const float* __restrict__ hsrc, const float* __restrict__ rinv_src,
    float* __restrict__ tmp, int E)
{
  long g = (long)blockIdx.x * 256 + threadIdx.x;
  int e = (int)(g >> 5);
  int lane = (int)(g & 31);
  if (e >= E) return;
  int s = src[e], d = dst[e];
  float sc = rinv_src[s];
  const float4 hv = *(const float4*)(hsrc + (long)s * 128 + lane * 4);
  float* tp = tmp + (long)d * 128 + lane * 4;
  atomicAdd(tp + 0, hv.x * sc);
  atomicAdd(tp + 1, hv.y * sc);
  atomicAdd(tp + 2, hv.z * sc);
  atomicAdd(tp + 3, hv.w * sc);
}

// Y[M x 128] (+)= ((rowscale ? A*rowscale : A)[M x K]) @ W[K x 128] + bias[128]
// Each wave owns a 16x32 output tile: one A fragment feeds TWO
// V_WMMA_F32_16X16X4_F32 (exact fp32), halving A-operand traffic per WMMA.
// 8 waves/block cover a 32-row x 128-col slab; grid = ceil(M/32).
__global__ __launch_bounds__(256) void gemm_wmma(
    const float* __restrict__ A, const float* __restrict__ W,
    const float* __restrict__ bias, const float* __restrict__ rowscale,
    float* __restrict__ Y, int M, int K, int accumulate, int act)
{
  const int wave = threadIdx.x >> 5;               // 0..7
  const int lane = threadIdx.x & 31;
  const int lh   = lane & 15;                      // 0..15
  const int hi   = lane >> 4;                      // half-wave: K pair (A/B), M half (C/D)
  const int m0   = (blockIdx.x * 2 + (wave >> 2)) * 16;
  const int n0   = (wave & 3) * 32;                // this wave: cols [n0, n0+32)

  int row = m0 + lh;
  if (row >= M) row = M - 1;                       // clamp; out-of-range rows never stored
  const float rs = rowscale ? rowscale[row] : 1.0f;
  const float* __restrict__ arow = A + (long)row * K;

  v8f c0 = {}, c1 = {};
  if (accumulate) {
#pragma unroll
    for (int v = 0; v < 8; ++v) {
      int rr = m0 + hi * 8 + v;
      if (rr < M) {
        c0[v] = Y[(long)rr * 128 + n0 + lh];
        c1[v] = Y[(long)rr * 128 + n0 + 16 + lh];
      }
    }
  }

  // A 16x4 fragment: lane lh holds row m0+lh; VGPR0/1 = K {0,1} (lanes 0-15) / {2,3} (lanes 16-31)
  // B 4x16 fragments: lane lh holds col n0+lh (resp. n0+16+lh); same K striping by half-wave.
#pragma unroll 4
  for (int k0 = 0; k0 < K; k0 += 4) {
    const int ka = k0 + hi * 2;
    v2f a;  a.x  = arow[ka] * rs;                        a.y  = arow[ka + 1] * rs;
    const float* w0 = W + (long)ka * 128 + n0 + lh;
    const float* w1 = W + (long)(ka + 1) * 128 + n0 + lh;
    v2f b0; b0.x = w0[0];   b0.y = w1[0];
    v2f b1; b1.x = w0[16];  b1.y = w1[16];
    c0 = __builtin_amdgcn_wmma_f32_16x16x4_f32(false, a, false, b0, (short)0, c0, false, false);
    c1 = __builtin_amdgcn_wmma_f32_16x16x4_f32(false, a, false, b1, (short)0, c1, false, false);
  }

  const float bv0 = bias ? bias[n0 + lh]      : 0.0f;
  const float bv1 = bias ? bias[n0 + 16 + lh] : 0.0f;
#pragma unroll
  for (int v = 0; v < 8; ++v) {
    int rr = m0 + hi * 8 + v;
    if (rr < M) {
      float v0 = c0[v] + bv0;
      float v1 = c1[v] + bv1;
      if (act) { v0 = lrelu(v0); v1 = lrelu(v1); }
      Y[(long)rr * 128 + n0 + lh]      = v0;
      Y[(long)rr * 128 + n0 + 16 + lh] = v1;
    }
  }
}

// p = lrelu(p * invR) elementwise (mean over relations + activation)
__global__ __launch_bounds__(256) void finalize_lrelu(float* __restrict__ p, long n, float invR) {
  long g = (long)blockIdx.x * 256 + threadIdx.x;
  if (g < n) p[g] = lrelu(p[g] * invR);
}

// out[r][0..1] = h[r][:] @ Wl(128x2) + bl
__global__ __launch_bounds__(256) void linear_out(
    const float* __restrict__ h, const float* __restrict__ Wl,
    const float* __restrict__ bl, float* __restrict__ out, int M)
{
  int r = blockIdx.x * 256 + threadIdx.x;
  if (r >= M) return;
  const float* __restrict__ hr = h + (long)r * 128;
  float a0 = bl[0], a1 = bl[1];
#pragma unroll 8
  for (int k = 0; k < 128; ++k) {
    float v = hr[k];
    a0 += v * Wl[2 * k + 0];
    a1 += v * Wl[2 * k + 1];
  }
  out[(long)r * 2 + 0] = a0;
  out[(long)r * 2 + 1] = a1;
}

// ---------------- host orchestration ----------------

extern "C" void kernel_launch(void* const* d_in, const int* in_sizes, int n_in,
                              void* d_out, int out_size, void* d_ws, size_t ws_size,
                              hipStream_t stream)
{
  (void)in_sizes; (void)n_in; (void)out_size; (void)ws_size;

  const float* x[4]   = {(const float*)d_in[0],  (const float*)d_in[1],
                         (const float*)d_in[2],  (const float*)d_in[3]};
  const float* Wi1[4] = {(const float*)d_in[4],  (const float*)d_in[6],
                         (const float*)d_in[8],  (const float*)d_in[10]};
  const float* bi1[4] = {(const float*)d_in[5],  (const float*)d_in[7],
                         (const float*)d_in[9],  (const float*)d_in[11]};
  const float* Wi2 = (const float*)d_in[12];
  const float* bi2 = (const float*)d_in[13];
  const float* c1W = (const float*)d_in[14];
  const float* c1b = (const float*)d_in[15];
  const float* c2W = (const float*)d_in[16];
  const float* c2b = (const float*)d_in[17];
  const float* Wl[4] = {(const float*)d_in[18], (const float*)d_in[20],
                        (const float*)d_in[22], (const float*)d_in[24]};
  const float* bl[4] = {(const float*)d_in[19], (const float*)d_in[21],
                        (const float*)d_in[23], (const float*)d_in[25]};
  const int* eidx[6] = {(const int*)d_in[26], (const int*)d_in[27], (const int*)d_in[28],
                        (const int*)d_in[29], (const int*)d_in[30], (const int*)d_in[31]};

  // node types: 0=user 1=news 2=source 3=follower
  const int  N_T[4]   = {200000, 100000, 5000, 200000};
  const long R_OFF[4] = {0, 200000, 300000, 305000};
  const int  IN_T[4]  = {128, 300, 128, 128};
  const long TOT_ROWS = 505000;

  // workspace: hA | hB | tmp | deg  (~623 MB total, fp32)
  float* hA  = (float*)d_ws;                 // 505000*128
  float* hB  = hA + TOT_ROWS * 128;          // 505000*128
  float* tmp = hB + TOT_ROWS * 128;          // 200000*128
  float* deg = tmp + (long)200000 * 128;     // 805000 (counts -> rsqrt degrees)

  float* out = (float*)d_out;
  const long L_OFF[4] = {0, 400000, 600000, 610000};   // logits (N_t x 2)
  float* h1 = out + 1010000;                            // h1 region, same row layout as hA/hB

  // ---- degrees (shared by conv1/conv2): histogram each of the 6 index arrays ----
  const long degN = 805000;
  zero_f32<<<(int)((degN + 255) / 256), 256, 0, stream>>>(deg, degN);
  const long DG_OFF[6] = {0, 200000, 300000, 305000, 405000, 605000};
  const int  DG_E[6]   = {500000, 500000, 100000, 100000, 1000000, 1000000};
  for (int a = 0; a < 6; ++a)
    count_edges<<<(DG_E[a] + 255) / 256, 256, 0, stream>>>(eidx[a], (int*)deg + DG_OFF[a], DG_E[a]);
  deg_to_rinv<<<(int)((degN + 255) / 256), 256, 0, stream>>>(deg, (int)degN);

  // ---- input layer 1: hB = lrelu(x @ Wi1 + bi1) ----
  for (int t = 0; t < 4; ++t) {
    int M = N_T[t];
    gemm_wmma<<<(M + 31) / 32, 256, 0, stream>>>(x[t], Wi1[t], bi1[t], nullptr,
                                                 hB + R_OFF[t] * 128, M, IN_T[t], 0, 1);
  }
  // ---- input layer 2: hA = lrelu(hB @ Wi2[t] + bi2[t]) ----
  for (int t = 0; t < 4; ++t) {
    int M = N_T[t];
    gemm_wmma<<<(M + 31) / 32, 256, 0, stream>>>(hB + R_OFF[t] * 128, Wi2 + (long)t * 16384,
                                                 bi2 + t * 128, nullptr,
                                                 hA + R_OFF[t] * 128, M, 128, 0, 1);
  }

  // relation metadata (r: 0 posts, 1 posted_by, 2 publishes, 3 published_by, 4 follows, 5 followed_by)
  const int  rel_srct[6] = {0, 1, 2, 1, 3, 0};
  const int  rel_dstt[6] = {1, 0, 1, 2, 0, 3};
  const int  rel_E[6]    = {500000, 500000, 100000, 100000, 1000000, 1000000};
  const int  rel_sarr[6] = {0, 1, 2, 3, 4, 5};   // index into eidx for src ids
  const int  rel_darr[6] = {1, 0, 3, 2, 5, 4};   // index into eidx for dst ids
  const long rel_soff[6] = {0, 200000, 300000, 305000, 405000, 605000};  // rinv(deg_out) offset
  const long rel_doff[6] = {200000, 0, 305000, 300000, 605000, 405000};  // rinv(deg_in)  offset
  const float invR[4] = {0.5f, 0.5f, 1.0f, 1.0f};  // 1 / len(DST_RELS[t])

  // ---- conv1: hA -> hB ; conv2: hB -> h1 (in d_out) ----
  for (int layer = 0; layer < 2; ++layer) {
    const float* hin  = (layer == 0) ? hA : hB;
    float*       hout = (layer == 0) ? hB : h1;
    const float* cw   = (layer == 0) ? c1W : c2W;
    const float* cb   = (layer == 0) ? c1b : c2b;

    const long hn = TOT_ROWS * 128;
    zero_f32<<<(int)((hn + 255) / 256), 256, 0, stream>>>(hout, hn);

    for (int r = 0; r < 6; ++r) {
      const int t = rel_dstt[r], s = rel_srct[r];
      const int Md = N_T[t];
      const long tn = (long)Md * 128;
      zero_f32<<<(int)((tn + 255) / 256), 256, 0, stream>>>(tmp, tn);

      const long threads = (long)rel_E[r] * 32;
      scatter_msgs<<<(int)((threads + 255) / 256), 256, 0, stream>>>(
          eidx[rel_sarr[r]], eidx[rel_darr[r]],
          hin + R_OFF[s] * 128, deg + rel_soff[r], tmp, rel_E[r]);

      // hout_t += (tmp * rinv_in) @ W[t][r] + b[t][r]
      gemm_wmma<<<(Md + 31) / 32, 256, 0, stream>>>(
          tmp, cw + ((long)t * 6 + r) * 16384, cb + ((long)t * 6 + r) * 128,
          deg + rel_doff[r], hout + R_OFF[t] * 128, Md, 128, 1, 0);
    }

    for (int t = 0; t < 4; ++t) {
      const long n = (long)N_T[t] * 128;
      finalize_lrelu<<<(int)((n + 255) / 256), 256, 0, stream>>>(hout + R_OFF[t] * 128, n, invR[t]);
    }
  }

  // ---- output head: logits = h1 @ Wl + bl ----
  for (int t = 0; t < 4; ++t) {
    int M = N_T[t];
    linear_out<<<(M + 255) / 256, 256, 0, stream>>>(h1 + R_OFF[t] * 128, Wl[t], bl[t],
                                                    out + L_OFF[t], M);
  }
}